// SAGPoolLayers_1692217115478
// MI455X (gfx1250) — compile-verified
//
#include <hip/hip_runtime.h>
#include <hip/hip_bf16.h>

// Problem constants (fixed by the reference file)
#define NHID 256
#define NPG  512
#define KKEEP 410

typedef __attribute__((ext_vector_type(2))) float v2f;
typedef __attribute__((ext_vector_type(8))) float v8f;

// ---------------------------------------------------------------------------
// init: deg = 0, score = 0, new_id = -1
// ---------------------------------------------------------------------------
__global__ void init_kernel(float* __restrict__ deg, float* __restrict__ score,
                            int* __restrict__ new_id, int n) {
    int i = blockIdx.x * blockDim.x + threadIdx.x;
    if (i < n) {
        deg[i]    = 0.0f;
        score[i]  = 0.0f;
        new_id[i] = -1;
    }
}

// ---------------------------------------------------------------------------
// deg[dst] += ew  (segment_sum over incoming edges)
// ---------------------------------------------------------------------------
__global__ void deg_kernel(const int* __restrict__ dst, const float* __restrict__ ew,
                           float* __restrict__ deg, int E) {
    int e = blockIdx.x * blockDim.x + threadIdx.x;
    if (e < E) atomicAdd(&deg[dst[e]], ew[e]);
}

// ---------------------------------------------------------------------------
// h = x @ W + b  via V_WMMA_F32_16X16X4_F32 (f32 in, f32 accumulate).
// One wave computes 16 rows. B broadcasts W[k..k+3] into all 16 columns, so
// every column of the 16x16 accumulator equals the matvec result for that row.
// C/D layout: VGPR j = row j (lanes 0-15) / row 8+j (lanes 16-31).
// A layout (32-bit 16x4): lanes 0-15 hold K=0,1; lanes 16-31 hold K=2,3.
// ---------------------------------------------------------------------------
__global__ __launch_bounds__(256) void matvec_wmma_kernel(
    const float* __restrict__ x, const float* __restrict__ W,
    const float* __restrict__ b, float* __restrict__ h, int n) {
    const int lane = threadIdx.x & 31;
    const int wave = threadIdx.x >> 5;
    const int tile = blockIdx.x * (blockDim.x >> 5) + wave;   // 16 rows per tile
    const int r0   = tile << 4;
    if (r0 >= n) return;                                      // wave-uniform branch

    const int m    = lane & 15;
    const int half = lane >> 4;                               // 0 or 1
    const float* __restrict__ xrow = x + (size_t)(r0 + m) * NHID;

    v8f c = {};
    #pragma unroll 4
    for (int k = 0; k < NHID; k += 4) {
        const int kk = k + half * 2;
        v2f a;  a.x  = xrow[kk];  a.y  = xrow[kk + 1];
        v2f bb; bb.x = W[kk];     bb.y = W[kk + 1];
        c = __builtin_amdgcn_wmma_f32_16x16x4_f32(
                /*neg_a=*/false, a, /*neg_b=*/false, bb,
                /*c_mod=*/(short)0, c, /*reuse_a=*/false, /*reuse_b=*/false);
    }

    const float bias = b[0];
    if (m == 0) {                       // lanes 0 and 16 hold rows 0-7 / 8-15
        const int base = r0 + half * 8;
        #pragma unroll
        for (int j = 0; j < 8; ++j) h[base + j] = c[j] + bias;
    }
}

// ---------------------------------------------------------------------------
// score[dst] += ew * rsqrt(deg[src]*deg[dst] + 1e-12) * h[src]
// ---------------------------------------------------------------------------
__global__ void score_scatter_kernel(const int* __restrict__ src, const int* __restrict__ dst,
                                     const float* __restrict__ ew, const float* __restrict__ deg,
                                     const float* __restrict__ h, float* __restrict__ score,
                                     int E) {
    int e = blockIdx.x * blockDim.x + threadIdx.x;
    if (e < E) {
        const int s = src[e], d = dst[e];
        const float nw = ew[e] * rsqrtf(deg[s] * deg[d] + 1e-12f);
        atomicAdd(&score[d], nw * h[s]);
    }
}

// ---------------------------------------------------------------------------
// Per-graph top-k via LDS bitonic sort of 512 (score,index) composite keys.
// Key = (~monotone(score) << 32) | index  -> ascending sort == descending
// score with lower-index-first tiebreak (matches jax.lax.top_k).
// Emits perm, new_id, batch_pool, and tanh(score[perm]) scale.
// ---------------------------------------------------------------------------
__global__ __launch_bounds__(256) void topk_kernel(
    const float* __restrict__ score, const int* __restrict__ batch,
    int* __restrict__ new_id, int* __restrict__ perm_out,
    int* __restrict__ batch_pool, float* __restrict__ scale) {
    __shared__ unsigned long long keys[NPG];
    const int g   = blockIdx.x;
    const int tid = threadIdx.x;
    const float* __restrict__ sc = score + (size_t)g * NPG;

    for (int i = tid; i < NPG; i += blockDim.x) {
        unsigned u = __float_as_uint(sc[i]);
        u = (u & 0x80000000u) ? ~u : (u | 0x80000000u);   // monotone ascending map
        keys[i] = ((unsigned long long)(~u) << 32) | (unsigned)i;
    }

    for (unsigned sz = 2; sz <= NPG; sz <<= 1) {
        for (unsigned stride = sz >> 1; stride > 0; stride >>= 1) {
            __syncthreads();
            const unsigned t  = tid;                                  // 256 pairs
            const unsigned lo = ((t & ~(stride - 1u)) << 1) | (t & (stride - 1u));
            const unsigned hi = lo | stride;
            const bool asc = ((lo & sz) == 0);
            const unsigned long long a = keys[lo], bq = keys[hi];
            if ((a > bq) == asc) { keys[lo] = bq; keys[hi] = a; }
        }
    }
    __syncthreads();

    for (int j = tid; j < KKEEP; j += blockDim.x) {
        const int local = (int)(keys[j] & 0xFFFFFFFFull);
        const int node  = g * NPG + local;
        const int out   = g * KKEEP + j;
        perm_out[out]   = node;
        new_id[node]    = out;
        batch_pool[out] = batch[node];
        scale[out]      = tanhf(sc[local]);
    }
}

// ---------------------------------------------------------------------------
// x_pool[row] = x[perm[row]] * scale[row]   (float4, 64 lanes per row)
// ---------------------------------------------------------------------------
__global__ __launch_bounds__(256) void gather_kernel(
    const float* __restrict__ x, const int* __restrict__ perm,
    const float* __restrict__ scale, float* __restrict__ xp, int nrows) {
    const int rowInBlk = threadIdx.x >> 6;
    const int c4       = threadIdx.x & 63;          // 64 x float4 = 256 floats
    const int row      = blockIdx.x * 4 + rowInBlk;
    if (row >= nrows) return;
    const int   node = perm[row];
    const float s    = scale[row];
    const float4 v = ((const float4*)(x + (size_t)node * NHID))[c4];
    float4 r; r.x = v.x * s; r.y = v.y * s; r.z = v.z * s; r.w = v.w * s;
    ((float4*)(xp + (size_t)row * NHID))[c4] = r;
}

// ---------------------------------------------------------------------------
// filter_adj: remap endpoints through new_id, mask dropped edges
// ---------------------------------------------------------------------------
__global__ void filter_kernel(const int* __restrict__ src, const int* __restrict__ dst,
                              const float* __restrict__ ew, const int* __restrict__ new_id,
                              int* __restrict__ ei0, int* __restrict__ ei1,
                              float* __restrict__ ea, int E) {
    int e = blockIdx.x * blockDim.x + threadIdx.x;
    if (e >= E) return;
    const int a = new_id[src[e]];
    const int b = new_id[dst[e]];
    const bool keep = (a >= 0) && (b >= 0);
    ei0[e] = keep ? a : -1;
    ei1[e] = keep ? b : -1;
    ea[e]  = keep ? ew[e] : 0.0f;
}

// ---------------------------------------------------------------------------
extern "C" void kernel_launch(void* const* d_in, const int* in_sizes, int n_in,
                              void* d_out, int out_size, void* d_ws, size_t ws_size,
                              hipStream_t stream) {
    const float* x   = (const float*)d_in[0];
    const float* W   = (const float*)d_in[1];
    const float* b   = (const float*)d_in[2];
    const int*   ei  = (const int*)d_in[3];
    const float* ew  = (const float*)d_in[4];
    const int*   bat = (const int*)d_in[5];

    const int N = in_sizes[0] / NHID;     // 131072
    const int E = in_sizes[4];            // 4194304
    const int G = N / NPG;                // 256
    const int GK = G * KKEEP;             // 104960

    const int* src = ei;
    const int* dst = ei + E;

    // workspace layout (4-byte units): deg | h | score | new_id | scale
    float* deg    = (float*)d_ws;
    float* h      = deg + N;
    float* score  = h + N;
    int*   new_id = (int*)(score + N);
    float* scale  = (float*)(new_id + N);

    // output layout: x_pool | ei0 | ei1 | ea | batch_pool | perm
    float* out   = (float*)d_out;
    float* xp    = out;
    int*   ei0   = (int*)(out + (size_t)GK * NHID);
    int*   ei1   = ei0 + E;
    float* ea    = (float*)(ei1 + E);
    int*   bpool = (int*)(ea + E);
    int*   permo = bpool + GK;

    const int TB = 256;
    init_kernel<<<(N + TB - 1) / TB, TB, 0, stream>>>(deg, score, new_id, N);
    deg_kernel<<<(E + TB - 1) / TB, TB, 0, stream>>>(dst, ew, deg, E);
    matvec_wmma_kernel<<<N / 128, TB, 0, stream>>>(x, W, b, h, N);
    score_scatter_kernel<<<(E + TB - 1) / TB, TB, 0, stream>>>(src, dst, ew, deg, h, score, E);
    topk_kernel<<<G, TB, 0, stream>>>(score, bat, new_id, permo, bpool, scale);
    gather_kernel<<<(GK + 3) / 4, TB, 0, stream>>>(x, permo, scale, xp, GK);
    filter_kernel<<<(E + TB - 1) / TB, TB, 0, stream>>>(src, dst, ew, new_id, ei0, ei1, ea, E);
}